// DetrHungarianMatcher_6820408066756
// MI455X (gfx1250) — compile-verified
//
#include <hip/hip_runtime.h>
#include <hip/hip_bf16.h>
#include <stdint.h>

// Problem dims (fixed by setup_inputs)
#define B_  16
#define Q_  900
#define C_  91
#define T_  300
#define NROW (B_ * Q_)   // 14400
#define NCOL (B_ * T_)   // 4800
#define CPAD 96          // padded class stride -> 384B rows, 16B aligned
#define ROWS_PER_BLK 16
#define NQUAD (NCOL / 4) // 1200 column quads

typedef float v4f __attribute__((ext_vector_type(4)));

// ---------------------------------------------------------------------------
// Main cost kernel (placed first so the disasm snippet shows it).
// Block = 256 threads x 4 columns each (1024 cols) x 16 rows.
// Stages 16 softmax rows (6 KB, contiguous) into LDS via CDNA5 async-to-LDS,
// then streams the [16 x 1024] output tile with non-temporal b128 stores.
// ---------------------------------------------------------------------------
__global__ __launch_bounds__(256) void cost_kernel(
    const float* __restrict__ prob, const float* __restrict__ rowtab,
    const float* __restrict__ coltab, float* __restrict__ out) {
  __shared__ float sprob[ROWS_PER_BLK * CPAD];  // 1536 f32 = 6144 B

  const int tid  = threadIdx.x;
  const int row0 = blockIdx.y * ROWS_PER_BLK;
  const int q    = blockIdx.x * 256 + tid;      // column quad id

  // ---- CDNA5 async global->LDS copy: 384 x 16B chunks (6144 B) ----
  uint32_t lds0 = (uint32_t)(uintptr_t)(&sprob[0]) + (uint32_t)tid * 16u;
  const char* gsrc = (const char*)(prob + (size_t)row0 * CPAD) + (size_t)tid * 16;
  {
    unsigned long long ga = (unsigned long long)(uintptr_t)gsrc;
    asm volatile("global_load_async_to_lds_b128 %0, %1, off"
                 :: "v"(lds0), "v"(ga) : "memory");
  }
  if (tid < 128) {  // remaining chunks 256..383
    unsigned long long ga = (unsigned long long)(uintptr_t)(gsrc + 256 * 16);
    uint32_t lds1 = lds0 + 256u * 16u;
    asm volatile("global_load_async_to_lds_b128 %0, %1, off"
                 :: "v"(lds1), "v"(ga) : "memory");
  }
  asm volatile("s_wait_asynccnt 0" ::: "memory");
  __syncthreads();

  if (q >= NQUAD) return;

  // ---- per-column constants for 4 columns (held in registers) ----
  v4f   c0[4], c1[4];
  float a2[4];
  int   cls[4];
  #pragma unroll
  for (int k = 0; k < 4; ++k) {
    const v4f* ct = (const v4f*)(coltab + (size_t)(q * 4 + k) * 12);
    c0[k] = ct[0];                 // cx,cy,w,h
    c1[k] = ct[1];                 // x0,y0,x1,y1
    v4f t = ct[2];                 // area2, cls, 0, 0
    a2[k]  = t[0];
    cls[k] = __float_as_int(t[1]);
  }

  float* orow = out + (size_t)row0 * NCOL + (size_t)q * 4;
  #pragma unroll 2
  for (int r = 0; r < ROWS_PER_BLK; ++r) {
    // per-row constants: uniform address -> scalar loads
    const float* rt = rowtab + (size_t)(row0 + r) * 12;
    const float pcx = rt[0], pcy = rt[1], pw = rt[2], ph = rt[3];
    const float px0 = rt[4], py0 = rt[5], px1 = rt[6], py1 = rt[7];
    const float area1 = rt[8];

    v4f res;
    #pragma unroll
    for (int k = 0; k < 4; ++k) {
      const float p = sprob[r * CPAD + cls[k]];  // ds_load_b32 gather

      // L1 cdist on (cx,cy,w,h)
      float l1 = fabsf(pcx - c0[k][0]) + fabsf(pcy - c0[k][1])
               + fabsf(pw  - c0[k][2]) + fabsf(ph  - c0[k][3]);

      // intersection / union
      float iw = fminf(px1, c1[k][2]) - fmaxf(px0, c1[k][0]);
      float ih = fminf(py1, c1[k][3]) - fmaxf(py0, c1[k][1]);
      iw = fmaxf(iw, 0.0f); ih = fmaxf(ih, 0.0f);
      const float inter = iw * ih;
      const float uni   = area1 + a2[k] - inter;

      // enclosing box area
      float ew = fmaxf(px1, c1[k][2]) - fminf(px0, c1[k][0]);
      float eh = fmaxf(py1, c1[k][3]) - fminf(py0, c1[k][1]);
      ew = fmaxf(ew, 0.0f); eh = fmaxf(eh, 0.0f);
      const float ae = ew * eh;

      // -giou = 1 - uni/ae - inter/uni = 1 - (uni^2 + inter*ae) * rcp(ae*uni)
      const float d  = ae * uni;
      float rc = __builtin_amdgcn_rcpf(d);
      rc = rc * (2.0f - d * rc);              // Newton step
      const float ngiou = 1.0f - (uni * uni + inter * ae) * rc;

      res[k] = l1 - p + ngiou;
    }
    __builtin_nontemporal_store(res, (v4f*)orow);
    orow += NCOL;
  }
}

// ---------------------------------------------------------------------------
// Kernel 1: per-row softmax (one wave32 per row) + per-row geometry table.
// prob:   [NROW][CPAD] f32 (pad classes 91..95 zeroed)
// rowtab: [NROW][12]  {cx,cy,w,h, x0,y0,x1,y1, area, pad...}
// ---------------------------------------------------------------------------
__global__ __launch_bounds__(256) void prep_rows_kernel(
    const float* __restrict__ logits, const float* __restrict__ pred,
    float* __restrict__ prob, float* __restrict__ rowtab) {
  const int lane = threadIdx.x & 31;
  const int row  = blockIdx.x * 8 + (threadIdx.x >> 5);  // 1800 blocks * 8 waves

  const float* lr = logits + (size_t)row * C_;
  float v0 = lr[lane];            // classes 0..31
  float v1 = lr[lane + 32];       // classes 32..63
  float v2 = (lane + 64 < C_) ? lr[lane + 64] : -3.4e38f;  // 64..90

  float m = fmaxf(v0, fmaxf(v1, v2));
  #pragma unroll
  for (int o = 16; o >= 1; o >>= 1) m = fmaxf(m, __shfl_xor(m, o, 32));

  float e0 = __expf(v0 - m);
  float e1 = __expf(v1 - m);
  float e2 = (lane + 64 < C_) ? __expf(v2 - m) : 0.0f;
  float s = e0 + e1 + e2;
  #pragma unroll
  for (int o = 16; o >= 1; o >>= 1) s += __shfl_xor(s, o, 32);

  float inv = __builtin_amdgcn_rcpf(s);
  inv = inv * (2.0f - s * inv);   // one Newton step

  float* pr = prob + (size_t)row * CPAD;
  pr[lane]      = e0 * inv;
  pr[lane + 32] = e1 * inv;
  pr[lane + 64] = (lane + 64 < C_) ? e2 * inv : 0.0f;  // also zeroes pad 91..95

  if (lane == 0) {
    float cx = pred[row * 4 + 0], cy = pred[row * 4 + 1];
    float w  = pred[row * 4 + 2], h  = pred[row * 4 + 3];
    float x0 = cx - 0.5f * w, y0 = cy - 0.5f * h;
    float x1 = cx + 0.5f * w, y1 = cy + 0.5f * h;
    float* rt = rowtab + (size_t)row * 12;
    rt[0] = cx; rt[1] = cy; rt[2] = w;  rt[3] = h;
    rt[4] = x0; rt[5] = y0; rt[6] = x1; rt[7] = y1;
    rt[8] = (x1 - x0) * (y1 - y0);
  }
}

// ---------------------------------------------------------------------------
// Kernel 2: per-column table.
// coltab: [NCOL][12] {cx,cy,w,h, x0,y0,x1,y1, area, cls(int bits), 0, 0}
// ---------------------------------------------------------------------------
__global__ __launch_bounds__(256) void prep_cols_kernel(
    const float* __restrict__ tboxes, const int* __restrict__ tids,
    float* __restrict__ coltab) {
  const int j = blockIdx.x * 256 + threadIdx.x;
  if (j >= NCOL) return;
  float cx = tboxes[j * 4 + 0], cy = tboxes[j * 4 + 1];
  float w  = tboxes[j * 4 + 2], h  = tboxes[j * 4 + 3];
  float x0 = cx - 0.5f * w, y0 = cy - 0.5f * h;
  float x1 = cx + 0.5f * w, y1 = cy + 0.5f * h;
  float* ct = coltab + (size_t)j * 12;
  ct[0] = cx; ct[1] = cy; ct[2] = w;  ct[3] = h;
  ct[4] = x0; ct[5] = y0; ct[6] = x1; ct[7] = y1;
  ct[8] = (x1 - x0) * (y1 - y0);
  ct[9] = __int_as_float(tids[j]);
  ct[10] = 0.0f; ct[11] = 0.0f;
}

// ---------------------------------------------------------------------------
extern "C" void kernel_launch(void* const* d_in, const int* in_sizes, int n_in,
                              void* d_out, int out_size, void* d_ws, size_t ws_size,
                              hipStream_t stream) {
  const float* logits = (const float*)d_in[0];   // [16,900,91]
  const float* pred   = (const float*)d_in[1];   // [16,900,4]
  const int*   tids   = (const int*)d_in[2];     // [16,300]
  const float* tboxes = (const float*)d_in[3];   // [16,300,4]
  float*       out    = (float*)d_out;           // [16,900,4800]

  char* ws = (char*)d_ws;
  float* prob   = (float*)ws;                                   // 14400*96*4 = 5.53 MB
  float* rowtab = (float*)(ws + (size_t)NROW * CPAD * 4);       // 14400*12*4 = 0.69 MB
  float* coltab = (float*)(ws + (size_t)NROW * CPAD * 4
                              + (size_t)NROW * 12 * 4);         // 4800*12*4  = 0.23 MB

  prep_rows_kernel<<<NROW / 8, 256, 0, stream>>>(logits, pred, prob, rowtab);
  prep_cols_kernel<<<(NCOL + 255) / 256, 256, 0, stream>>>(tboxes, tids, coltab);

  dim3 grid((NQUAD + 255) / 256, NROW / ROWS_PER_BLK);          // (5, 900)
  cost_kernel<<<grid, 256, 0, stream>>>(prob, rowtab, coltab, out);
}